// ReaxffBondEnergy_24404004175957
// MI455X (gfx1250) — compile-verified
//
#include <hip/hip_runtime.h>
#include <hip/hip_bf16.h>

#ifndef __has_builtin
#define __has_builtin(x) 0
#endif

#define LOG2E_F 1.44269504088896340736f

// Pointer/vector types for the gfx1250 async global->LDS builtin:
// signature (probe-confirmed): (global int4* src, shared int4* dst, imm, imm)
typedef int v4i_t __attribute__((ext_vector_type(4)));
typedef __attribute__((address_space(1))) v4i_t* global_v4i_ptr;
typedef __attribute__((address_space(3))) v4i_t* lds_v4i_ptr;

// ---------------------------------------------------------------------------
// CDNA5 hardware transcendentals: v_log_f32 / v_exp_f32 are base-2.
// ---------------------------------------------------------------------------
__device__ __forceinline__ float amd_log2(float x) {
#if __has_builtin(__builtin_amdgcn_logf)
    return __builtin_amdgcn_logf(x);          // v_log_f32 (log2)
#else
    float r; asm("v_log_f32 %0, %1" : "=v"(r) : "v"(x)); return r;
#endif
}
__device__ __forceinline__ float amd_exp2(float x) {
#if __has_builtin(__builtin_amdgcn_exp2f)
    return __builtin_amdgcn_exp2f(x);         // v_exp_f32 (exp2)
#else
    float r; asm("v_exp_f32 %0, %1" : "=v"(r) : "v"(x)); return r;
#endif
}

// No-return global_atomic_add_f32 (resolves in L2 atomic units).
__device__ __forceinline__ void atomic_add_f32(float* addr, float v) {
#if defined(__HIP_PLATFORM_AMD__) || defined(__HIP__)
    unsafeAtomicAdd(addr, v);
#else
    atomicAdd(addr, v);
#endif
}

// ---------------------------------------------------------------------------
// Kernel 1: zero the accumulator (harness poisons d_out with 0xAA).
// ---------------------------------------------------------------------------
__global__ __launch_bounds__(256) void zero_f32(float* __restrict__ p, int n) {
    int i = blockIdx.x * blockDim.x + threadIdx.x;
    if (i < n) p[i] = 0.0f;
}

// ---------------------------------------------------------------------------
// Per-bond energy given padded LDS param row (8 floats per type, 16B-aligned
// rows => ds_load_b128 + ds_load_b32 for the 5 live params).
// ---------------------------------------------------------------------------
__device__ __forceinline__ float bond_energy_one(const float* __restrict__ P8,
                                                 int t, float bs, float bp, float bpp) {
    const float* p = P8 + (t << 3);
    float4 d  = *(const float4*)p;     // de_s, de_p, de_pp, p_be1  (ds_load_b128)
    float pbe2 = p[4];                 // (ds_load_b32)
    // bs^pbe2 = exp2(pbe2 * log2(bs));  bs in (0.05, 1) so log2 is safe
    float l2bs = amd_log2(bs);
    float tpow = amd_exp2(pbe2 * l2bs);
    // exp(p_be1*(1-tpow)) = exp2(LOG2E * p_be1 * (1-tpow))
    float ex   = amd_exp2(LOG2E_F * d.w * (1.0f - tpow));
    return -d.x * bs * ex - d.y * bp - d.z * bpp;
}

// ---------------------------------------------------------------------------
// Kernel 2: bond energy + segment-sum via f32 atomics.
// Each thread handles 4 bonds via B128 loads. The 16x5 param table is staged
// into LDS with gfx1250 async global->LDS B128 copies, then repacked to a
// 16x8 padded layout for vectorized LDS gathers.
// ---------------------------------------------------------------------------
__global__ __launch_bounds__(256) void reaxff_bond_energy_kernel(
    const int*   __restrict__ bond_type,
    const int*   __restrict__ atom_i,
    const float* __restrict__ bo_sigma,
    const float* __restrict__ bo_pi,
    const float* __restrict__ bo_pipi,
    const float* __restrict__ bond_params,   // 16 x 5 floats, row-major
    float*       __restrict__ e_atom,
    int n_bonds)
{
    __shared__ __align__(16) float Praw[80];      // raw 16x5 staging
    __shared__ __align__(16) float P8[16 * 8];    // padded 16x8 table

#if __has_builtin(__builtin_amdgcn_global_load_async_to_lds_b128)
    // Async DMA of the 80-float table: lanes 0..19 each move 16 bytes.
    if (threadIdx.x < 20) {
        __builtin_amdgcn_global_load_async_to_lds_b128(
            (global_v4i_ptr)(bond_params + threadIdx.x * 4),
            (lds_v4i_ptr)(&Praw[threadIdx.x * 4]),
            0, 0);
    }
#if __has_builtin(__builtin_amdgcn_s_wait_asynccnt)
    __builtin_amdgcn_s_wait_asynccnt(0);
#else
    asm volatile("s_wait_asynccnt 0" ::: "memory");
#endif
    __syncthreads();
#else
    if (threadIdx.x < 80) Praw[threadIdx.x] = bond_params[threadIdx.x];
    __syncthreads();
#endif

    // Repack 16x5 -> 16x8 (padded) so per-bond gathers vectorize.
    if (threadIdx.x < 16) {
#pragma unroll
        for (int j = 0; j < 5; ++j)
            P8[threadIdx.x * 8 + j] = Praw[threadIdx.x * 5 + j];
    }
    __syncthreads();

    int tid  = blockIdx.x * blockDim.x + threadIdx.x;
    int base = tid * 4;
    if (base + 3 < n_bonds) {
        // Fast path: B128 loads for everything.
        float4 bs4  = ((const float4*)bo_sigma)[tid];
        float4 bp4  = ((const float4*)bo_pi)[tid];
        float4 bpp4 = ((const float4*)bo_pipi)[tid];
        int4   bt4  = ((const int4*)bond_type)[tid];
        int4   ai4  = ((const int4*)atom_i)[tid];

        float bs[4]  = {bs4.x,  bs4.y,  bs4.z,  bs4.w};
        float bp[4]  = {bp4.x,  bp4.y,  bp4.z,  bp4.w};
        float bpp[4] = {bpp4.x, bpp4.y, bpp4.z, bpp4.w};
        int   bt[4]  = {bt4.x,  bt4.y,  bt4.z,  bt4.w};
        int   ai[4]  = {ai4.x,  ai4.y,  ai4.z,  ai4.w};

#pragma unroll
        for (int k = 0; k < 4; ++k) {
            float e = bond_energy_one(P8, bt[k], bs[k], bp[k], bpp[k]);
            atomic_add_f32(&e_atom[ai[k]], e);
        }
    } else if (base < n_bonds) {
        // Tail (not hit for N_BONDS = 2^24, kept for generality).
        for (int b = base; b < n_bonds; ++b) {
            float e = bond_energy_one(P8, bond_type[b], bo_sigma[b], bo_pi[b], bo_pipi[b]);
            atomic_add_f32(&e_atom[atom_i[b]], e);
        }
    }
}

// ---------------------------------------------------------------------------
// Host-side launcher (graph-capture safe: only kernel launches on `stream`).
// Input order: bond_type, atom_i, bo_sigma, bo_pi, bo_pipi, bond_params, n_atoms
// ---------------------------------------------------------------------------
extern "C" void kernel_launch(void* const* d_in, const int* in_sizes, int n_in,
                              void* d_out, int out_size, void* d_ws, size_t ws_size,
                              hipStream_t stream) {
    const int*   bond_type   = (const int*)  d_in[0];
    const int*   atom_i      = (const int*)  d_in[1];
    const float* bo_sigma    = (const float*)d_in[2];
    const float* bo_pi       = (const float*)d_in[3];
    const float* bo_pipi     = (const float*)d_in[4];
    const float* bond_params = (const float*)d_in[5];
    float*       e_atom      = (float*)d_out;

    int n_bonds = in_sizes[0];
    int n_atoms = out_size;

    // 1) zero the accumulator
    {
        int blocks = (n_atoms + 255) / 256;
        zero_f32<<<blocks, 256, 0, stream>>>(e_atom, n_atoms);
    }
    // 2) bond energy + atomic segment sum (4 bonds per thread)
    {
        int nthreads = (n_bonds + 3) / 4;
        int blocks   = (nthreads + 255) / 256;
        reaxff_bond_energy_kernel<<<blocks, 256, 0, stream>>>(
            bond_type, atom_i, bo_sigma, bo_pi, bo_pipi, bond_params,
            e_atom, n_bonds);
    }
}